// Affine_AE_2D_module_30751965839997
// MI455X (gfx1250) — compile-verified
//
#include <hip/hip_runtime.h>
#include <math.h>

#define Bn   64
#define Cn   8
#define Hn   256
#define Wn   256
#define EMBn 64
#define LATn 32
#define HWn  (Hn * Wn)        // 65536
#define IMGn (Cn * HWn)       // 524288
#define NPIXn (Bn * HWn)      // 4194304
#define NTOTn (Bn * IMGn)     // 33554432

typedef __attribute__((ext_vector_type(2))) float v2f;
typedef __attribute__((ext_vector_type(8))) float v8f;

// ---------------------------------------------------------------- bilinear
struct Bilin { int o00, o01, o10, o11; float w00, w01, w10, w11; };

__device__ __forceinline__ Bilin make_bilin(const float* __restrict__ t, int x, int y) {
    float xs = (x + 0.5f) * (2.0f / Wn) - 1.0f;
    float ys = (y + 0.5f) * (2.0f / Hn) - 1.0f;
    float gx = t[0] * xs + t[1] * ys + t[2];
    float gy = t[3] * xs + t[4] * ys + t[5];
    float ix = ((gx + 1.0f) * Wn - 1.0f) * 0.5f;
    float iy = ((gy + 1.0f) * Hn - 1.0f) * 0.5f;
    float x0f = floorf(ix), y0f = floorf(iy);
    int x0 = (int)x0f, y0 = (int)y0f;
    int x1 = x0 + 1,  y1 = y0 + 1;
    float wx1 = ix - x0f, wy1 = iy - y0f;
    float wx0 = 1.0f - wx1, wy0 = 1.0f - wy1;
    bool vx0 = (x0 >= 0) && (x0 < Wn);
    bool vx1 = (x1 >= 0) && (x1 < Wn);
    bool vy0 = (y0 >= 0) && (y0 < Hn);
    bool vy1 = (y1 >= 0) && (y1 < Hn);
    int xc0 = min(max(x0, 0), Wn - 1), xc1 = min(max(x1, 0), Wn - 1);
    int yc0 = min(max(y0, 0), Hn - 1), yc1 = min(max(y1, 0), Hn - 1);
    Bilin r;
    r.o00 = yc0 * Wn + xc0;  r.o01 = yc0 * Wn + xc1;
    r.o10 = yc1 * Wn + xc0;  r.o11 = yc1 * Wn + xc1;
    r.w00 = (vx0 && vy0) ? wy0 * wx0 : 0.0f;
    r.w01 = (vx1 && vy0) ? wy0 * wx1 : 0.0f;
    r.w10 = (vx0 && vy1) ? wy1 * wx0 : 0.0f;
    r.w11 = (vx1 && vy1) ? wy1 * wx1 : 0.0f;
    return r;
}

// ---------------------------------------------------------------- kernels
__global__ void __launch_bounds__(256) k_zero(float* __restrict__ p, int n) {
    int i = blockIdx.x * 256 + threadIdx.x;
    if (i < n) p[i] = 0.0f;
}

// one block per (b,c): mean over H*W
__global__ void __launch_bounds__(256) k_mean(const float* __restrict__ x,
                                              float* __restrict__ means) {
    int bc = blockIdx.x;
    const float* p = x + (size_t)bc * HWn;
    float s = 0.0f;
    for (int i = threadIdx.x; i < HWn; i += 256) s += p[i];
    __shared__ float sm[256];
    sm[threadIdx.x] = s;
    __syncthreads();
    for (int off = 128; off > 0; off >>= 1) {
        if (threadIdx.x < off) sm[threadIdx.x] += sm[threadIdx.x + off];
        __syncthreads();
    }
    if (threadIdx.x == 0) means[bc] = sm[0] * (1.0f / HWn);
}

// emb_affine[64,64] = means[64,8] @ W_ae[8,64] via v_wmma_f32_16x16x4_f32
// 16 waves (512 threads): wave (mi,ni) tile, 2 k-steps of 4.
__global__ void __launch_bounds__(512) k_emb_affine(const float* __restrict__ means,
                                                    const float* __restrict__ W_ae,
                                                    float* __restrict__ emb) {
    int wave = threadIdx.x >> 5;
    int lane = threadIdx.x & 31;
    int mi = wave >> 2, ni = wave & 3;
    int mrow  = mi * 16 + (lane & 15);
    int ncol  = ni * 16 + (lane & 15);
    int khalf = (lane >> 4) * 2;           // 0 for lanes 0-15, 2 for 16-31
    v8f c = {};
#pragma unroll
    for (int ks = 0; ks < 2; ++ks) {
        int kb = ks * 4 + khalf;
        v2f a, b;
        a.x = means[mrow * Cn + kb];
        a.y = means[mrow * Cn + kb + 1];
        b.x = W_ae[(kb)     * EMBn + ncol];
        b.y = W_ae[(kb + 1) * EMBn + ncol];
        c = __builtin_amdgcn_wmma_f32_16x16x4_f32(false, a, false, b,
                                                  (short)0, c, false, false);
    }
    int mbase = mi * 16 + ((lane >> 4) ? 8 : 0);
#pragma unroll
    for (int r = 0; r < 8; ++r)
        emb[(mbase + r) * EMBn + ncol] = c[r];
}

// thetas (pass order rot,scale,shear,trans) into ws; output order into d_out tail.
__global__ void __launch_bounds__(64) k_heads(const float* __restrict__ emb,
                                              const float* __restrict__ W_rot,
                                              const float* __restrict__ W_scale,
                                              const float* __restrict__ W_shear,
                                              const float* __restrict__ W_trans,
                                              const float* __restrict__ W_mask,
                                              float* __restrict__ thetas,   // [4][64*6]
                                              float* __restrict__ maskp,    // [64]
                                              float* __restrict__ out_tail) // scale,shear,rot,trans,mask
{
    int b = threadIdx.x;
    if (b >= Bn) return;
    const float* e = emb + b * EMBn;
    const float* Wh[4] = { W_rot, W_scale, W_shear, W_trans };  // pass order
    const int   slot[4] = { 2, 0, 1, 3 };                       // output order slots
    const float eye[6] = { 1.f, 0.f, 0.f, 0.f, 1.f, 0.f };
    for (int h = 0; h < 4; ++h) {
        float acc[6] = { 0, 0, 0, 0, 0, 0 };
        for (int k = 0; k < EMBn; ++k) {
            float ev = e[k];
            const float* wr = Wh[h] + k * 6;
#pragma unroll
            for (int j = 0; j < 6; ++j) acc[j] += ev * wr[j];
        }
#pragma unroll
        for (int j = 0; j < 6; ++j) {
            float v = acc[j] * 0.1f + eye[j];
            thetas[h * (Bn * 6) + b * 6 + j]          = v;
            out_tail[slot[h] * (Bn * 6) + b * 6 + j]  = v;
        }
    }
    float m = 0.0f;
    for (int k = 0; k < EMBn; ++k) m += e[k] * W_mask[k];
    float sg = 1.0f / (1.0f + __expf(-m));
    maskp[b] = sg;
    out_tail[4 * (Bn * 6) + b] = sg;
}

// one bilinear warp pass; each block = 256 contiguous pixels of one batch image.
__global__ void __launch_bounds__(256) k_warp(const float* __restrict__ src,
                                              float* __restrict__ dst,
                                              const float* __restrict__ thetas,
                                              int pass) {
    int b   = blockIdx.x >> 8;                       // 256 blocks per batch
    int pix = ((blockIdx.x & 255) << 8) + threadIdx.x;
    __shared__ float ts[6];
    if (threadIdx.x < 6) ts[threadIdx.x] = thetas[pass * (Bn * 6) + b * 6 + threadIdx.x];
    __syncthreads();
    Bilin g = make_bilin(ts, pix & 255, pix >> 8);
    size_t base = (size_t)b * IMGn;
#pragma unroll
    for (int c = 0; c < Cn; ++c) {
        const float* p = src + base + c * HWn;
        float v = p[g.o00] * g.w00 + p[g.o01] * g.w01 +
                  p[g.o10] * g.w10 + p[g.o11] * g.w11;
        dst[base + c * HWn + pix] = v;
    }
}

// fused: warp pass 4 (translation) + mask + 8x8 decode + per-channel sums.
__global__ void __launch_bounds__(256) k_final(const float* __restrict__ src,
                                               float* __restrict__ dec_out,
                                               const float* __restrict__ thetas,
                                               const float* __restrict__ maskp,
                                               const float* __restrict__ W_dec,
                                               float* __restrict__ sums) {
    int b   = blockIdx.x >> 8;
    int pix = ((blockIdx.x & 255) << 8) + threadIdx.x;
    __shared__ float ts[6];
    __shared__ float wd[64];
    if (threadIdx.x < 6)  ts[threadIdx.x] = thetas[3 * (Bn * 6) + b * 6 + threadIdx.x];
    if (threadIdx.x < 64) wd[threadIdx.x] = W_dec[threadIdx.x];
    __syncthreads();
    Bilin g = make_bilin(ts, pix & 255, pix >> 8);
    float mk = maskp[b];
    size_t base = (size_t)b * IMGn;
    float xt[Cn];
#pragma unroll
    for (int c = 0; c < Cn; ++c) {
        const float* p = src + base + c * HWn;
        xt[c] = (p[g.o00] * g.w00 + p[g.o01] * g.w01 +
                 p[g.o10] * g.w10 + p[g.o11] * g.w11) * mk;
    }
#pragma unroll
    for (int d = 0; d < Cn; ++d) {
        float acc = 0.0f;
#pragma unroll
        for (int c = 0; c < Cn; ++c) acc += xt[c] * wd[c * Cn + d];
        dec_out[base + d * HWn + pix] = acc;
    }
    // per-channel partial sums -> global atomics (for emb)
    __shared__ float sm[256];
    for (int c = 0; c < Cn; ++c) {
        sm[threadIdx.x] = xt[c];
        __syncthreads();
        for (int off = 128; off > 0; off >>= 1) {
            if (threadIdx.x < off) sm[threadIdx.x] += sm[threadIdx.x + off];
            __syncthreads();
        }
        if (threadIdx.x == 0) atomicAdd(&sums[b * Cn + c], sm[0]);
        __syncthreads();
    }
}

// emb[64,32] = (sums/HW)[64,8] @ W_enc[8,32] via v_wmma_f32_16x16x4_f32 (8 waves)
__global__ void __launch_bounds__(256) k_emb2(const float* __restrict__ sums,
                                              const float* __restrict__ W_enc,
                                              float* __restrict__ emb_out) {
    int wave = threadIdx.x >> 5;
    int lane = threadIdx.x & 31;
    int mi = wave >> 1, ni = wave & 1;
    int mrow  = mi * 16 + (lane & 15);
    int ncol  = ni * 16 + (lane & 15);
    int khalf = (lane >> 4) * 2;
    const float scl = 1.0f / HWn;
    v8f c = {};
#pragma unroll
    for (int ks = 0; ks < 2; ++ks) {
        int kb = ks * 4 + khalf;
        v2f a, b;
        a.x = sums[mrow * Cn + kb]     * scl;
        a.y = sums[mrow * Cn + kb + 1] * scl;
        b.x = W_enc[(kb)     * LATn + ncol];
        b.y = W_enc[(kb + 1) * LATn + ncol];
        c = __builtin_amdgcn_wmma_f32_16x16x4_f32(false, a, false, b,
                                                  (short)0, c, false, false);
    }
    int mbase = mi * 16 + ((lane >> 4) ? 8 : 0);
#pragma unroll
    for (int r = 0; r < 8; ++r)
        emb_out[(mbase + r) * LATn + ncol] = c[r];
}

// ---------------------------------------------------------------- launch
extern "C" void kernel_launch(void* const* d_in, const int* in_sizes, int n_in,
                              void* d_out, int out_size, void* d_ws, size_t ws_size,
                              hipStream_t stream) {
    const float* x       = (const float*)d_in[0];
    const float* W_ae    = (const float*)d_in[1];
    const float* W_rot   = (const float*)d_in[2];
    const float* W_scale = (const float*)d_in[3];
    const float* W_shear = (const float*)d_in[4];
    const float* W_trans = (const float*)d_in[5];
    const float* W_mask  = (const float*)d_in[6];
    const float* W_enc   = (const float*)d_in[7];
    const float* W_dec   = (const float*)d_in[8];

    float* out   = (float*)d_out;
    float* dec   = out;                               // [64,8,256,256]
    float* emb_o = out + (size_t)NTOTn;               // [64,32]
    float* tail  = emb_o + Bn * LATn;                 // scale,shear,rot,trans,mask

    float* ws     = (float*)d_ws;
    float* bufA   = ws;                               // NTOTn floats (ping-pong)
    float* means1 = bufA + (size_t)NTOTn;             // 512
    float* embA   = means1 + Bn * Cn;                 // 4096
    float* thetas = embA + Bn * EMBn;                 // 4*384
    float* maskp  = thetas + 4 * Bn * 6;              // 64
    float* sums   = maskp + Bn;                       // 512

    const int WARP_BLOCKS = NPIXn / 256;              // 16384

    k_zero<<<2, 256, 0, stream>>>(sums, Bn * Cn);
    k_mean<<<Bn * Cn, 256, 0, stream>>>(x, means1);
    k_emb_affine<<<1, 512, 0, stream>>>(means1, W_ae, embA);
    k_heads<<<1, 64, 0, stream>>>(embA, W_rot, W_scale, W_shear, W_trans, W_mask,
                                  thetas, maskp, tail);
    k_warp<<<WARP_BLOCKS, 256, 0, stream>>>(x,    bufA, thetas, 0);  // rotation
    k_warp<<<WARP_BLOCKS, 256, 0, stream>>>(bufA, dec,  thetas, 1);  // scale
    k_warp<<<WARP_BLOCKS, 256, 0, stream>>>(dec,  bufA, thetas, 2);  // shear
    k_final<<<WARP_BLOCKS, 256, 0, stream>>>(bufA, dec, thetas, maskp, W_dec, sums);
    k_emb2<<<1, 256, 0, stream>>>(sums, W_enc, emb_o);
}